// RestGCNEqualHidden_22539988369859
// MI455X (gfx1250) — compile-verified
//
#include <hip/hip_runtime.h>
#include <hip/hip_bf16.h>

typedef __attribute__((ext_vector_type(2))) float v2f;
typedef __attribute__((ext_vector_type(8))) float v8f;

// ---------------------------------------------------------------------------
// Elementwise helpers
// ---------------------------------------------------------------------------
__global__ __launch_bounds__(256)
void k_fill_f32(float* __restrict__ p, float v, long long n) {
  long long i = (long long)blockIdx.x * blockDim.x + threadIdx.x;
  if (i < n) p[i] = v;
}

__global__ __launch_bounds__(256)
void k_deg_accum(const long long* __restrict__ dst, float* __restrict__ deg, long long E) {
  long long i = (long long)blockIdx.x * blockDim.x + threadIdx.x;
  if (i < E) unsafeAtomicAdd(&deg[dst[i]], 1.0f);
}

__global__ __launch_bounds__(256)
void k_rsqrt_inplace(float* __restrict__ p, long long n) {
  long long i = (long long)blockIdx.x * blockDim.x + threadIdx.x;
  if (i < n) {
    float d = p[i];
    p[i] = (d > 0.0f) ? rsqrtf(d) : 0.0f;
  }
}

// Wt[c, k] = W[k, c]  (K x M row-major  ->  M x K row-major)
__global__ __launch_bounds__(256)
void k_transpose(const float* __restrict__ W, float* __restrict__ Wt, int K, int M) {
  long long i = (long long)blockIdx.x * blockDim.x + threadIdx.x;
  if (i >= (long long)K * M) return;
  int k = (int)(i / M);
  int c = (int)(i - (long long)k * M);
  Wt[(long long)c * K + k] = W[i];
}

// out[row, f] = h[row, f] * dinv[row]^2 + bias[f]  (self-loop msg + bias)
// grid = (ceil(Fdim/64), N), block = 64
__global__ __launch_bounds__(64)
void k_selfloop_bias(const float* __restrict__ h, const float* __restrict__ dinv,
                     const float* __restrict__ bias, float* __restrict__ out, int Fdim) {
  const int f = blockIdx.x * 64 + threadIdx.x;
  if (f >= Fdim) return;
  const long long row = blockIdx.y;
  const float d = dinv[row];
  const long long idx = row * Fdim + f;
  out[idx] = h[idx] * d * d + bias[f];
}

// x2 = relu(agg) + x
__global__ __launch_bounds__(256)
void k_relu_residual(const float* __restrict__ agg, const float* __restrict__ x,
                     float* __restrict__ x2, long long total) {
  long long i = (long long)blockIdx.x * blockDim.x + threadIdx.x;
  if (i < total) x2[i] = fmaxf(agg[i], 0.0f) + x[i];
}

// ---------------------------------------------------------------------------
// Edge scatter: one wave per edge, float4 gather, fp32 HW atomics (L2-resident).
// Fdim must be a multiple of 4 (128 and 40 both are).
// ---------------------------------------------------------------------------
__global__ __launch_bounds__(256)
void k_edge_scatter(const long long* __restrict__ src, const long long* __restrict__ dst,
                    const float* __restrict__ h, const float* __restrict__ dinv,
                    float* __restrict__ out, long long E, int Fdim) {
  const int lane = threadIdx.x & 31;
  const long long e = (long long)blockIdx.x * (blockDim.x >> 5) + (threadIdx.x >> 5);
  if (e >= E) return;
  const long long s = src[e];
  const long long d = dst[e];
  const float nrm = dinv[s] * dinv[d];
  const float* __restrict__ hp = h + s * (long long)Fdim;
  float* __restrict__ op = out + d * (long long)Fdim;
  for (int f = lane * 4; f < Fdim; f += 128) {
    const float4 v = *(const float4*)(hp + f);
    unsafeAtomicAdd(op + f + 0, v.x * nrm);
    unsafeAtomicAdd(op + f + 1, v.y * nrm);
    unsafeAtomicAdd(op + f + 2, v.z * nrm);
    unsafeAtomicAdd(op + f + 3, v.w * nrm);
  }
}

// ---------------------------------------------------------------------------
// fp32 WMMA GEMM: C[N,M] = A[N,K] @ Bt^T, with Bt = B transposed (M x K).
// One wave computes a 16x64 strip: 4 accumulators, 4 back-to-back
// V_WMMA_F32_16X16X4_F32 per K-step.
//
// Software-pipelined: each iteration issues the NEXT K-step's 5 global_load_b64
// (index-based -> immediate offsets, no per-step address math), then runs the
// 4 independent WMMAs on the CURRENT operands, then rotates registers.
// The wait before a WMMA group therefore only needs loadcnt<=5 (next step
// in flight), never a full drain -> one K-step of load latency hidden.
//
// FULL=true strips skip all column predication. EXEC stays all-ones
// (WMMA requirement); partial tiles use value-selects for loads and
// predicate only the final stores.
//
// ISA VGPR layouts (CDNA5 ISA):
//   A 16x4 : lanes 0-15 row=l K=(k,k+1); lanes 16-31 row=l K=(k+2,k+3)
//   B  4x16: lanes 0-15 col=l K=(k,k+1); lanes 16-31 col=l K=(k+2,k+3)
//   C/D    : VGPR i, half h -> row = 8*h + i, col = l
// ---------------------------------------------------------------------------
template <bool FULL>
__global__ __launch_bounds__(32)
void k_gemm_wmma_f32(const float* __restrict__ A, const float* __restrict__ Bt,
                     float* __restrict__ C, int N, int K, int M) {
  const int lane = threadIdx.x;
  const int half = lane >> 4;   // 0 or 1 -> K sub-pair
  const int l    = lane & 15;
  const int row0 = blockIdx.y << 4;
  const int col0 = blockIdx.x << 6;   // 64-wide column strip

  const int arow = row0 + l;
  const int ar   = (arow < N) ? arow : 0;   // clamp; OOB-row values never stored

  const int c0 = col0 + 0 * 16 + l;
  const int c1 = col0 + 1 * 16 + l;
  const int c2 = col0 + 2 * 16 + l;
  const int c3 = col0 + 3 * 16 + l;
  const bool v0 = FULL || (c0 < M);
  const bool v1 = FULL || (c1 < M);
  const bool v2 = FULL || (c2 < M);
  const bool v3 = FULL || (c3 < M);
  const float s0 = v0 ? 1.0f : 0.0f;
  const float s1 = v1 ? 1.0f : 0.0f;
  const float s2 = v2 ? 1.0f : 0.0f;
  const float s3 = v3 ? 1.0f : 0.0f;

  // Fixed base pointers; all K-steps use immediate offsets off these.
  const float* __restrict__ ap  = A  + (long long)ar * K + half * 2;
  const float* __restrict__ bp0 = Bt + (long long)(v0 ? c0 : 0) * K + half * 2;
  const float* __restrict__ bp1 = Bt + (long long)(v1 ? c1 : 0) * K + half * 2;
  const float* __restrict__ bp2 = Bt + (long long)(v2 ? c2 : 0) * K + half * 2;
  const float* __restrict__ bp3 = Bt + (long long)(v3 ? c3 : 0) * K + half * 2;

  v8f acc0 = {}, acc1 = {}, acc2 = {}, acc3 = {};

  // ---- prologue: load K-step 0 ----
  v2f a, b0, b1, b2, b3;
  a.x  = ap[0];  a.y  = ap[1];
  b0.x = bp0[0]; b0.y = bp0[1];
  b1.x = bp1[0]; b1.y = bp1[1];
  b2.x = bp2[0]; b2.y = bp2[1];
  b3.x = bp3[0]; b3.y = bp3[1];
  if (!FULL) { b0 *= s0; b1 *= s1; b2 *= s2; b3 *= s3; }

  // ---- steady state: issue loads for step k, compute step k-4 ----
  for (int k = 4; k < K; k += 4) {
    v2f an, bn0, bn1, bn2, bn3;
    an.x  = ap[k];      an.y  = ap[k + 1];
    bn0.x = bp0[k];     bn0.y = bp0[k + 1];
    bn1.x = bp1[k];     bn1.y = bp1[k + 1];
    bn2.x = bp2[k];     bn2.y = bp2[k + 1];
    bn3.x = bp3[k];     bn3.y = bp3[k + 1];
    if (!FULL) { bn0 *= s0; bn1 *= s1; bn2 *= s2; bn3 *= s3; }

    acc0 = __builtin_amdgcn_wmma_f32_16x16x4_f32(false, a, false, b0, (short)0, acc0, false, false);
    acc1 = __builtin_amdgcn_wmma_f32_16x16x4_f32(false, a, false, b1, (short)0, acc1, false, false);
    acc2 = __builtin_amdgcn_wmma_f32_16x16x4_f32(false, a, false, b2, (short)0, acc2, false, false);
    acc3 = __builtin_amdgcn_wmma_f32_16x16x4_f32(false, a, false, b3, (short)0, acc3, false, false);

    a = an; b0 = bn0; b1 = bn1; b2 = bn2; b3 = bn3;
  }

  // ---- epilogue: last K-step ----
  acc0 = __builtin_amdgcn_wmma_f32_16x16x4_f32(false, a, false, b0, (short)0, acc0, false, false);
  acc1 = __builtin_amdgcn_wmma_f32_16x16x4_f32(false, a, false, b1, (short)0, acc1, false, false);
  acc2 = __builtin_amdgcn_wmma_f32_16x16x4_f32(false, a, false, b2, (short)0, acc2, false, false);
  acc3 = __builtin_amdgcn_wmma_f32_16x16x4_f32(false, a, false, b3, (short)0, acc3, false, false);

  const int rbase = row0 + half * 8;
#pragma unroll
  for (int i = 0; i < 8; ++i) {
    const int r = rbase + i;
    if (r < N) {
      float* __restrict__ cp = C + (long long)r * M;
      if (FULL || (c0 < M)) cp[c0] = acc0[i];
      if (FULL || (c1 < M)) cp[c1] = acc1[i];
      if (FULL || (c2 < M)) cp[c2] = acc2[i];
      if (FULL || (c3 < M)) cp[c3] = acc3[i];
    }
  }
}

// ---------------------------------------------------------------------------
// Launch
// ---------------------------------------------------------------------------
extern "C" void kernel_launch(void* const* d_in, const int* in_sizes, int n_in,
                              void* d_out, int out_size, void* d_ws, size_t ws_size,
                              hipStream_t stream) {
  const float*     x  = (const float*)d_in[0];
  const long long* ei = (const long long*)d_in[1];   // int64 edge_index [2,E]
  const float*     W1 = (const float*)d_in[2];
  const float*     b1 = (const float*)d_in[3];
  const float*     W2 = (const float*)d_in[4];
  const float*     b2 = (const float*)d_in[5];

  const int       Hd = in_sizes[3];                  // hidden = 128
  const int       F  = in_sizes[2] / Hd;             // in feats = 128
  const long long N  = (long long)in_sizes[0] / F;   // 50000
  const long long E  = (long long)in_sizes[1] / 2;   // 800000
  const int       Cd = in_sizes[5];                  // classes = 40

  const long long* src = ei;
  const long long* dst = ei + E;

  // Workspace partition (fp32, 16B-aligned slices)
  float* ws   = (float*)d_ws;
  float* dinv = ws;                                   // N
  float* h1   = dinv + ((N + 3) & ~3LL);              // N*Hd  (reused as x2)
  float* out1 = h1 + (long long)N * Hd;               // N*Hd
  float* h2   = out1 + (long long)N * Hd;             // N*Cd
  float* W1t  = h2 + (((long long)N * Cd + 3) & ~3LL);// Hd*F  (W1 transposed)
  float* W2t  = W1t + (long long)F * Hd;              // Cd*Hd (W2 transposed)
  float* y    = (float*)d_out;                        // N*Cd

  const int TB = 256;
  const unsigned gN   = (unsigned)((N + TB - 1) / TB);
  const unsigned gE   = (unsigned)((E + TB - 1) / TB);
  const unsigned gNH  = (unsigned)((N * Hd + TB - 1) / TB);
  const unsigned gEsc = (unsigned)((E + 7) / 8);      // 8 waves (edges) per block

  // 0) transpose weights so WMMA B operands load contiguous K (b64 loads)
  k_transpose<<<(unsigned)(((long long)F * Hd + TB - 1) / TB), TB, 0, stream>>>(W1, W1t, F, Hd);
  k_transpose<<<(unsigned)(((long long)Hd * Cd + TB - 1) / TB), TB, 0, stream>>>(W2, W2t, Hd, Cd);

  // 1) degree (self-loop contributes 1), then dinv = rsqrt(deg)
  k_fill_f32<<<gN, TB, 0, stream>>>(dinv, 1.0f, N);
  k_deg_accum<<<gE, TB, 0, stream>>>(dst, dinv, E);
  k_rsqrt_inplace<<<gN, TB, 0, stream>>>(dinv, N);

  // 2) h1 = x @ W1  (fp32 WMMA, 16x64 strips, Hd=128 -> FULL strips)
  {
    dim3 grid((Hd + 63) / 64, (unsigned)((N + 15) / 16));
    if (Hd % 64 == 0)
      k_gemm_wmma_f32<true><<<grid, 32, 0, stream>>>(x, W1t, h1, (int)N, F, Hd);
    else
      k_gemm_wmma_f32<false><<<grid, 32, 0, stream>>>(x, W1t, h1, (int)N, F, Hd);
  }

  // 3) aggregate layer 1: out1 = b1 + dinv^2*h1 (self loop) + edge messages
  {
    dim3 grid((Hd + 63) / 64, (unsigned)N);
    k_selfloop_bias<<<grid, 64, 0, stream>>>(h1, dinv, b1, out1, Hd);
  }
  k_edge_scatter<<<gEsc, TB, 0, stream>>>(src, dst, h1, dinv, out1, E, Hd);

  // 4) x2 = relu(out1) + x  (overwrite h1 buffer)
  k_relu_residual<<<gNH, TB, 0, stream>>>(out1, x, h1, N * Hd);

  // 5) h2 = x2 @ W2  (fp32 WMMA, Cd=40 -> predicated strip)
  {
    dim3 grid((Cd + 63) / 64, (unsigned)((N + 15) / 16));
    if (Cd % 64 == 0)
      k_gemm_wmma_f32<true><<<grid, 32, 0, stream>>>(h1, W2t, h2, (int)N, Hd, Cd);
    else
      k_gemm_wmma_f32<false><<<grid, 32, 0, stream>>>(h1, W2t, h2, (int)N, Hd, Cd);
  }

  // 6) aggregate layer 2 directly into d_out
  {
    dim3 grid((Cd + 63) / 64, (unsigned)N);
    k_selfloop_bias<<<grid, 64, 0, stream>>>(h2, dinv, b2, y, Cd);
  }
  k_edge_scatter<<<gEsc, TB, 0, stream>>>(src, dst, h2, dinv, y, E, Cd);
}